// PostProcessor_62732292326155
// MI455X (gfx1250) — compile-verified
//
#include <hip/hip_runtime.h>
#include <hip/hip_bf16.h>
#include <math.h>

// ---------------- problem constants (from reference) ----------------
#define B_   64
#define C_   3
#define H_   96
#define W_   320
#define HW_  (H_ * W_)        // 30720
#define REG_ 8
#define KDET 100
#define PI_F 3.14159265358979323846f
#define DEPTH_REF0 28.01f
#define DEPTH_REF1 16.32f
#define DET_THRESH 0.25f

typedef __attribute__((ext_vector_type(4))) unsigned int v4u;
typedef __attribute__((ext_vector_type(8))) int          v8i;
typedef __attribute__((ext_vector_type(4))) int          v4i;

// =====================================================================
// Kernel 1: per-(b,c) top-100 over the 30720-element heatmap slice.
// The whole slice (120 KB) is DMA'd into LDS with the CDNA5 Tensor Data
// Mover, then 100 rounds of block-wide argmax run entirely out of LDS.
// =====================================================================
__global__ __launch_bounds__(512) void topk_slice_kernel(
    const float* __restrict__ heat,
    float* __restrict__ s_all,   // [B*C*KDET]
    int*   __restrict__ i_all)   // [B*C*KDET]
{
    extern __shared__ float smem[];
    float* sheat = smem;                 // [HW_] slice copy (at dynamic-LDS base)
    float* rval  = smem + HW_;           // [512] reduction values
    int*   ridx  = (int*)(rval + 512);   // [512] reduction indices

    const int slice = blockIdx.x;                  // b*C + c
    const int tid   = threadIdx.x;
    const float* src = heat + (size_t)slice * HW_;

    // ---- TDM: async tensor load of the whole slice into LDS ----
    // One wave issues the DMA (TDM ignores EXEC, so wave-guard it).
    if (tid < 32) {
        unsigned lds_base = __builtin_amdgcn_groupstaticsize(); // dynamic LDS start
        unsigned long long ga = (unsigned long long)(uintptr_t)src;

        v4u g0;
        g0.x = 1u;                                        // count=1 (valid), user mode
        g0.y = lds_base;                                  // lds_addr (bytes)
        g0.z = (unsigned)(ga & 0xffffffffull);            // global_addr[31:0]
        g0.w = (unsigned)((ga >> 32) & 0x01ffffffull)     // global_addr[56:32]
             | (2u << 30);                                // type = 2 ("image")

        v8i g1;
        g1[0] = (int)(2u << 16);                          // data_size = 4B
        g1[1] = (int)(((unsigned)HW_ & 0xffffu) << 16);   // tensor_dim0[15:0]
        g1[2] = (int)(((unsigned)HW_ >> 16) | (1u << 16)); // tensor_dim0[31:16] | tensor_dim1=1
        g1[3] = (int)(((unsigned)HW_ & 0xffffu) << 16);   // tile_dim0 = 30720
        g1[4] = 1;                                        // tile_dim1 = 1, tile_dim2 = 0
        g1[5] = HW_;                                      // tensor_dim0_stride[31:0]
        g1[6] = (int)(((unsigned)HW_ & 0xffffu) << 16);   // dim0_stride hi=0 | dim1_stride lo
        g1[7] = 0;

        v4i z4 = {0, 0, 0, 0};                            // groups 2/3 unused (<=2D)
        v8i z8 = {0, 0, 0, 0, 0, 0, 0, 0};
        __builtin_amdgcn_tensor_load_to_lds(g0, g1, z4, z4, z8, 0);
        __builtin_amdgcn_s_wait_tensorcnt(0);             // s_wait_tensorcnt 0
    }
    __syncthreads();

    // ---- 100 iterations of block-wide argmax (lowest index wins ties) ----
    for (int it = 0; it < KDET; ++it) {
        float bv = -__builtin_inff();
        int   bi = 0;
        for (int i = tid; i < HW_; i += 512) {
            float v = sheat[i];
            if (v > bv) { bv = v; bi = i; }               // strict '>' => lowest idx per thread
        }
        rval[tid] = bv; ridx[tid] = bi;
        __syncthreads();
        for (int s = 256; s > 0; s >>= 1) {
            if (tid < s) {
                float ov = rval[tid + s]; int oi = ridx[tid + s];
                if (ov > rval[tid] || (ov == rval[tid] && oi < ridx[tid])) {
                    rval[tid] = ov; ridx[tid] = oi;
                }
            }
            __syncthreads();
        }
        if (tid == 0) {
            s_all[slice * KDET + it] = rval[0];
            i_all[slice * KDET + it] = ridx[0];
            sheat[ridx[0]] = -__builtin_inff();           // knock out the winner
        }
        __syncthreads();
    }
}

// ---------------- 3x3 inverse (adjugate) ----------------
__device__ inline void inv3(const float* m, float* o) {
    float a = m[0], b = m[1], c = m[2];
    float d = m[3], e = m[4], f = m[5];
    float g = m[6], h = m[7], i = m[8];
    float A =  (e * i - f * h);
    float Bc = -(d * i - f * g);
    float Cc =  (d * h - e * g);
    float det = a * A + b * Bc + c * Cc;
    float inv = 1.0f / det;
    o[0] = A * inv;              o[1] = -(b * i - c * h) * inv;  o[2] =  (b * f - c * e) * inv;
    o[3] = Bc * inv;             o[4] =  (a * i - c * g) * inv;  o[5] = -(a * f - c * d) * inv;
    o[6] = Cc * inv;             o[7] = -(a * h - b * g) * inv;  o[8] =  (a * e - b * d) * inv;
}

// =====================================================================
// Kernel 2: per-batch second top-100 (over 300 class-major scores) +
// full per-detection geometry, one block per batch image.
// =====================================================================
__global__ __launch_bounds__(128) void detect_kernel(
    const float* __restrict__ reg,      // [B,8,H,W]
    const float* __restrict__ trans,    // [B,3,3]
    const float* __restrict__ Kmat,     // [B,3,3]
    const float* __restrict__ Ksrc,     // [B,3,3]
    const float* __restrict__ size2,    // [B,2]
    const float* __restrict__ s_all,    // [B*C*KDET]
    const int*   __restrict__ i_all,    // [B*C*KDET]
    float* __restrict__ out)            // [B*KDET, 14]
{
    __shared__ float vals[C_ * KDET];   // 300 scores (destroyed by selection)
    __shared__ float svv[128];
    __shared__ int   svi[128];
    __shared__ float selv[KDET];
    __shared__ int   seli[KDET];
    __shared__ float Tinv[9], Kinv[9], Ks[9];

    const int b = blockIdx.x, tid = threadIdx.x;

    for (int i = tid; i < C_ * KDET; i += 128) vals[i] = s_all[b * C_ * KDET + i];
    if (tid == 0) {
        inv3(trans + b * 9, Tinv);
        inv3(Kmat + b * 9, Kinv);
        for (int i = 0; i < 9; ++i) Ks[i] = Ksrc[b * 9 + i];
    }
    __syncthreads();

    // ---- top-100 of the 300 class-major scores ----
    for (int it = 0; it < KDET; ++it) {
        float bv = -__builtin_inff();
        int   bi = 0;
        for (int i = tid; i < C_ * KDET; i += 128) {
            float v = vals[i];
            if (v > bv) { bv = v; bi = i; }
        }
        svv[tid] = bv; svi[tid] = bi;
        __syncthreads();
        for (int s = 64; s > 0; s >>= 1) {
            if (tid < s) {
                float ov = svv[tid + s]; int oi = svi[tid + s];
                if (ov > svv[tid] || (ov == svv[tid] && oi < svi[tid])) {
                    svv[tid] = ov; svi[tid] = oi;
                }
            }
            __syncthreads();
        }
        if (tid == 0) {
            selv[it] = svv[0]; seli[it] = svi[0];
            vals[svi[0]] = -__builtin_inff();
        }
        __syncthreads();
    }

    // ---- per-detection geometry (threads 0..99) ----
    if (tid < KDET) {
        const float score = selv[tid];
        const int   j     = seli[tid];          // index into (C*K) class-major
        const int   cls   = j / KDET;
        const int   r     = j - cls * KDET;
        const int   ind   = i_all[(b * C_ + cls) * KDET + r];
        const float ysf   = (float)(ind / W_);
        const float xsf   = (float)(ind % W_);

        // gather 8 regression channels at (b, ch, ind)
        const float* rb = reg + (size_t)b * REG_ * HW_ + ind;
        float p0 = rb[0 * HW_], p1 = rb[1 * HW_], p2 = rb[2 * HW_], p3 = rb[3 * HW_];
        float p4 = rb[4 * HW_], p5 = rb[5 * HW_], p6 = rb[6 * HW_], p7 = rb[7 * HW_];

        const float depth = p0 * DEPTH_REF1 + DEPTH_REF0;

        // img_pts = trans_inv @ [x+dx, y+dy, 1] * depth ; locs = K_inv @ img_pts
        const float e0 = xsf + p1, e1 = ysf + p2, e2 = 1.0f;
        float i0 = (Tinv[0] * e0 + Tinv[1] * e1 + Tinv[2] * e2) * depth;
        float i1 = (Tinv[3] * e0 + Tinv[4] * e1 + Tinv[5] * e2) * depth;
        float i2 = (Tinv[6] * e0 + Tinv[7] * e1 + Tinv[8] * e2) * depth;
        float l0 = Kinv[0] * i0 + Kinv[1] * i1 + Kinv[2] * i2;
        float l1 = Kinv[3] * i0 + Kinv[4] * i1 + Kinv[5] * i2;
        float l2 = Kinv[6] * i0 + Kinv[7] * i1 + Kinv[8] * i2;

        const float DR[3][3] = {{3.88f, 1.63f, 1.53f},
                                {1.78f, 1.70f, 0.58f},
                                {0.88f, 0.67f, 1.77f}};
        const float d0 = expf(p3) * DR[cls][0];
        const float d1 = expf(p4) * DR[cls][1];
        const float d2 = expf(p5) * DR[cls][2];
        l1 += d1 * 0.5f;

        const float ray = atanf(l0 / (l2 + 1e-7f));
        float alpha = atanf(p6 / (p7 + 1e-7f));
        alpha += (p7 >= 0.0f) ? -(0.5f * PI_F) : (0.5f * PI_F);
        float roty = alpha + ray;
        if (roty  >  PI_F) roty  -= 2.0f * PI_F;
        if (roty  < -PI_F) roty  += 2.0f * PI_F;
        if (alpha >  PI_F) alpha -= 2.0f * PI_F;
        if (alpha < -PI_F) alpha += 2.0f * PI_F;

        // 8 box corners -> rotate -> translate -> project with K_src
        const float cr = cosf(roty), sr = sinf(roty);
        const float half[8] = {0.5f, 0.5f, 0.5f, 0.5f, -0.5f, -0.5f, -0.5f, -0.5f};
        const float down[8] = {0.f, 0.f, 0.f, 0.f, -1.f, -1.f, -1.f, -1.f};
        const int idx0[8] = {4, 0, 1, 2, 3, 5, 6, 7};
        const int idx1[8] = {4, 5, 0, 1, 6, 7, 2, 3};
        const int idx2[8] = {4, 5, 6, 0, 1, 2, 3, 7};

        float xmin =  __builtin_inff(), xmax = -__builtin_inff();
        float ymin =  __builtin_inff(), ymax = -__builtin_inff();
        #pragma unroll
        for (int k = 0; k < 8; ++k) {
            float cx = d0 * half[idx0[k]];
            float cy = d1 * down[idx1[k]];
            float cz = d2 * half[idx2[k]];
            float px =  cr * cx + sr * cz + l0;
            float py =  cy + l1;
            float pz = -sr * cx + cr * cz + l2;
            float u0 = Ks[0] * px + Ks[1] * py + Ks[2] * pz;
            float u1 = Ks[3] * px + Ks[4] * py + Ks[5] * pz;
            float u2 = Ks[6] * px + Ks[7] * py + Ks[8] * pz;
            float u = u0 / u2, v = u1 / u2;
            xmin = fminf(xmin, u); xmax = fmaxf(xmax, u);
            ymin = fminf(ymin, v); ymax = fmaxf(ymax, v);
        }
        const float wimg = size2[0], himg = size2[1];    // size[0,0], size[0,1]
        xmin = fminf(fmaxf(xmin, 0.0f), wimg);
        xmax = fminf(fmaxf(xmax, 0.0f), wimg);
        ymin = fminf(fmaxf(ymin, 0.0f), himg);
        ymax = fminf(fmaxf(ymax, 0.0f), himg);

        const bool keep = score > DET_THRESH;
        float row[14] = {(float)cls, alpha, xmin, ymin, xmax, ymax,
                         d1, d2, d0,              // roll(dims, -1)
                         l0, l1, l2, roty, score};
        float* o = out + ((size_t)b * KDET + tid) * 14;
        #pragma unroll
        for (int k = 0; k < 14; ++k) o[k] = keep ? row[k] : 0.0f;
    }
}

// =====================================================================
extern "C" void kernel_launch(void* const* d_in, const int* in_sizes, int n_in,
                              void* d_out, int out_size, void* d_ws, size_t ws_size,
                              hipStream_t stream) {
    const float* heat  = (const float*)d_in[0];   // (64,3,96,320)
    const float* reg   = (const float*)d_in[1];   // (64,8,96,320)
    const float* trans = (const float*)d_in[2];   // (64,3,3)
    const float* Kmat  = (const float*)d_in[3];   // (64,3,3)
    const float* Ksrc  = (const float*)d_in[4];   // (64,3,3)
    const float* size2 = (const float*)d_in[5];   // (64,2)
    float* out = (float*)d_out;                   // (6400,14)

    float* s_all = (float*)d_ws;
    int*   i_all = (int*)((char*)d_ws + (size_t)B_ * C_ * KDET * sizeof(float));

    const size_t shmem = (size_t)HW_ * sizeof(float) + 512 * sizeof(float) + 512 * sizeof(int);
    // 124 KB dynamic LDS (WGP has 320 KB) — raise the per-kernel cap.
    (void)hipFuncSetAttribute((const void*)topk_slice_kernel,
                              hipFuncAttributeMaxDynamicSharedMemorySize, (int)shmem);

    topk_slice_kernel<<<B_ * C_, 512, shmem, stream>>>(heat, s_all, i_all);
    detect_kernel<<<B_, 128, 0, stream>>>(reg, trans, Kmat, Ksrc, size2,
                                          s_all, i_all, out);
}